// RobustFeatureExtractorGNN_60559038873713
// MI455X (gfx1250) — compile-verified
//
#include <hip/hip_runtime.h>
#include <math.h>

// ---------------- problem constants ----------------
#define N_NODES 50000
#define E_EDGES 800000
#define E2      (E_EDGES + N_NODES)      // edges + self loops = 850000
#define IN_C    3
#define HID     128
#define OUT_C   64
#define DEPTH   4
#define HEADS   4
#define EPSLN   1e-6f
#define NEG_SLOPE 0.2f

typedef float v2f __attribute__((ext_vector_type(2)));
typedef float v8f __attribute__((ext_vector_type(8)));

// ---------------- helpers ----------------
__device__ __forceinline__ float waveSum(float v) {
  #pragma unroll
  for (int off = 16; off > 0; off >>= 1) v += __shfl_xor(v, off, 32);
  return v;
}

// float atomic max via int/uint monotonic-bits trick (init must be -inf)
__device__ __forceinline__ void atomicMaxF32(float* addr, float val) {
  if (!(val < 0.0f)) {
    atomicMax(reinterpret_cast<int*>(addr), __float_as_int(val));
  } else {
    atomicMin(reinterpret_cast<unsigned int*>(addr), __float_as_uint(val));
  }
}

// ---------------- kernels ----------------

// h = x @ W_in + b_in   (K = 3, pure VALU)
__global__ void k_in_proj(const float* __restrict__ x, const float* __restrict__ Win,
                          const float* __restrict__ bin, float* __restrict__ h) {
  int tid = blockIdx.x * blockDim.x + threadIdx.x;
  if (tid >= N_NODES * HID) return;
  int n = tid >> 7, c = tid & (HID - 1);
  float acc = bin[c];
  #pragma unroll
  for (int k = 0; k < IN_C; ++k) acc += x[n * IN_C + k] * Win[k * HID + c];
  h[tid] = acc;
}

// LayerNorm over last dim (128). One wave per row, float4 per lane.
__global__ void k_layernorm(const float* __restrict__ h, float* __restrict__ o,
                            const float* __restrict__ gamma, const float* __restrict__ beta) {
  int wave = (blockIdx.x * blockDim.x + threadIdx.x) >> 5;
  int lane = threadIdx.x & 31;
  if (wave >= N_NODES) return;
  const float4 v = *reinterpret_cast<const float4*>(h + (size_t)wave * HID + lane * 4);
  float s = v.x + v.y + v.z + v.w;
  s = waveSum(s);
  float mu = s * (1.0f / HID);
  float dx = v.x - mu, dy = v.y - mu, dz = v.z - mu, dw = v.w - mu;
  float q = dx * dx + dy * dy + dz * dz + dw * dw;
  q = waveSum(q);
  float inv = rsqrtf(q * (1.0f / HID) + EPSLN);
  const float4 g = *reinterpret_cast<const float4*>(gamma + lane * 4);
  const float4 b = *reinterpret_cast<const float4*>(beta + lane * 4);
  float4 r;
  r.x = dx * inv * g.x + b.x;
  r.y = dy * inv * g.y + b.y;
  r.z = dz * inv * g.z + b.z;
  r.w = dw * inv * g.w + b.w;
  *reinterpret_cast<float4*>(o + (size_t)wave * HID + lane * 4) = r;
}

// C[M x Ntot] = A[M x 128] @ B[128 x Ntot] (+ bias per column).
// Block = 256 threads (8 waves). blockIdx.y selects a 64-column slice staged in LDS.
// Each wave computes a 16x64 output tile with v_wmma_f32_16x16x4_f32.
__global__ void k_gemm_wmma(const float* __restrict__ A, const float* __restrict__ B,
                            float* __restrict__ C, const float* __restrict__ bias,
                            int Mtiles, int Ntot) {
  __shared__ float Bs[128 * 68];          // 68-float row stride: 16B aligned, bank-shifted
  const int n0  = blockIdx.y * 64;
  const int tid = threadIdx.x;

  // cooperative load of B[0..127][n0..n0+63] into LDS
  for (int i = tid; i < 128 * 16; i += 256) {
    int row = i >> 4, c4 = (i & 15) << 2;
    float4 v = *reinterpret_cast<const float4*>(B + (size_t)row * Ntot + n0 + c4);
    *reinterpret_cast<float4*>(&Bs[row * 68 + c4]) = v;
  }
  __syncthreads();

  const int wave = tid >> 5;
  const int lane = tid & 31;
  const int mt   = blockIdx.x * 8 + wave;          // wave-uniform -> EXEC stays all-ones
  if (mt >= Mtiles) return;

  const int m0    = mt * 16;
  const int lrow  = lane & 15;
  const int khalf = (lane >> 4) << 1;              // 0 or 2
  const float* __restrict__ Arow = A + (size_t)(m0 + lrow) * 128;

  v8f acc0 = {}, acc1 = {}, acc2 = {}, acc3 = {};
  #pragma unroll 4
  for (int k0 = 0; k0 < 128; k0 += 4) {
    v2f a;
    a.x = Arow[k0 + khalf];
    a.y = Arow[k0 + khalf + 1];
    const float* Bk = &Bs[(k0 + khalf) * 68];
    v2f b0, b1, b2, b3;
    b0.x = Bk[lrow];        b0.y = Bk[68 + lrow];
    b1.x = Bk[16 + lrow];   b1.y = Bk[68 + 16 + lrow];
    b2.x = Bk[32 + lrow];   b2.y = Bk[68 + 32 + lrow];
    b3.x = Bk[48 + lrow];   b3.y = Bk[68 + 48 + lrow];
    acc0 = __builtin_amdgcn_wmma_f32_16x16x4_f32(false, a, false, b0, (short)0, acc0, false, false);
    acc1 = __builtin_amdgcn_wmma_f32_16x16x4_f32(false, a, false, b1, (short)0, acc1, false, false);
    acc2 = __builtin_amdgcn_wmma_f32_16x16x4_f32(false, a, false, b2, (short)0, acc2, false, false);
    acc3 = __builtin_amdgcn_wmma_f32_16x16x4_f32(false, a, false, b3, (short)0, acc3, false, false);
  }

  const int rbase = m0 + ((lane >> 4) << 3);       // M = j (lanes 0-15) or 8+j (lanes 16-31)
  float bv0 = bias ? bias[n0 + lrow]      : 0.0f;
  float bv1 = bias ? bias[n0 + 16 + lrow] : 0.0f;
  float bv2 = bias ? bias[n0 + 32 + lrow] : 0.0f;
  float bv3 = bias ? bias[n0 + 48 + lrow] : 0.0f;
  #pragma unroll
  for (int j = 0; j < 8; ++j) {
    float* Cr = C + (size_t)(rbase + j) * Ntot + n0;
    Cr[lrow]      = acc0[j] + bv0;
    Cr[16 + lrow] = acc1[j] + bv1;
    Cr[32 + lrow] = acc2[j] + bv2;
    Cr[48 + lrow] = acc3[j] + bv3;
  }
}

// alpha_s/alpha_d: per-(node,head) dot(xh[n,h,:], att[h,:]). One wave each.
__global__ void k_alpha(const float* __restrict__ xh, const float* __restrict__ a_src,
                        const float* __restrict__ a_dst,
                        float* __restrict__ as_, float* __restrict__ ad_) {
  int wave = (blockIdx.x * blockDim.x + threadIdx.x) >> 5;
  int lane = threadIdx.x & 31;
  if (wave >= N_NODES * HEADS) return;
  int h = wave & (HEADS - 1);
  const float4 xv = *reinterpret_cast<const float4*>(xh + (size_t)wave * HID + lane * 4);
  const float4 sv = *reinterpret_cast<const float4*>(a_src + h * HID + lane * 4);
  const float4 dv = *reinterpret_cast<const float4*>(a_dst + h * HID + lane * 4);
  float s = xv.x * sv.x + xv.y * sv.y + xv.z * sv.z + xv.w * sv.w;
  float d = xv.x * dv.x + xv.y * dv.y + xv.z * dv.z + xv.w * dv.w;
  s = waveSum(s); d = waveSum(d);
  if (lane == 0) { as_[wave] = s; ad_[wave] = d; }
}

// init agg=0, m=-inf, z=0
__global__ void k_init(float* __restrict__ agg, float* __restrict__ m_, float* __restrict__ z_) {
  int tid = blockIdx.x * blockDim.x + threadIdx.x;
  if (tid < N_NODES * HEADS * HID) agg[tid] = 0.0f;
  if (tid < N_NODES * HEADS) {
    reinterpret_cast<unsigned int*>(m_)[tid] = 0xFF800000u;   // -inf
    z_[tid] = 0.0f;
  }
}

// pass 1: e = leaky_relu(alpha_s[src] + alpha_d[dst]); segment max via atomics
__global__ void k_edge_pass1(const int* __restrict__ ei, const float* __restrict__ as_,
                             const float* __restrict__ ad_, float* __restrict__ ew,
                             float* __restrict__ m_) {
  int tid = blockIdx.x * blockDim.x + threadIdx.x;
  if (tid >= E2 * HEADS) return;
  int e = tid >> 2, h = tid & (HEADS - 1);
  int src = (e < E_EDGES) ? ei[e] : (e - E_EDGES);
  int dst = (e < E_EDGES) ? ei[E_EDGES + e] : (e - E_EDGES);
  float v = as_[src * HEADS + h] + ad_[dst * HEADS + h];
  v = (v > 0.0f) ? v : NEG_SLOPE * v;
  ew[tid] = v;
  atomicMaxF32(&m_[dst * HEADS + h], v);
}

// pass 2: p = exp(e - m[dst]); segment sum; store p in place
__global__ void k_edge_pass2(const int* __restrict__ ei, float* __restrict__ ew,
                             const float* __restrict__ m_, float* __restrict__ z_) {
  int tid = blockIdx.x * blockDim.x + threadIdx.x;
  if (tid >= E2 * HEADS) return;
  int e = tid >> 2, h = tid & (HEADS - 1);
  int dst = (e < E_EDGES) ? ei[E_EDGES + e] : (e - E_EDGES);
  float p = __expf(ew[tid] - m_[dst * HEADS + h]);
  ew[tid] = p;
  atomicAdd(&z_[dst * HEADS + h], p);
}

// aggregation: one wave per (edge, head); lanes cover 128 channels as float4
__global__ void k_edge_agg(const int* __restrict__ ei, const float* __restrict__ ew,
                           const float* __restrict__ z_, const float* __restrict__ xh,
                           float* __restrict__ agg) {
  int wave = (blockIdx.x * blockDim.x + threadIdx.x) >> 5;
  int lane = threadIdx.x & 31;
  if (wave >= E2 * HEADS) return;
  int e = wave >> 2, h = wave & (HEADS - 1);
  int src = (e < E_EDGES) ? ei[e] : (e - E_EDGES);
  int dst = (e < E_EDGES) ? ei[E_EDGES + e] : (e - E_EDGES);
  float al = ew[wave] / z_[dst * HEADS + h];
  const float4 xv = *reinterpret_cast<const float4*>(xh + (size_t)(src * HEADS + h) * HID + lane * 4);
  float* a = agg + (size_t)(dst * HEADS + h) * HID + lane * 4;
  atomicAdd(a + 0, al * xv.x);
  atomicAdd(a + 1, al * xv.y);
  atomicAdd(a + 2, al * xv.z);
  atomicAdd(a + 3, al * xv.w);
}

// head mean + bias, relu, + residual
__global__ void k_finalize(const float* __restrict__ agg, const float* __restrict__ bconv,
                           const float* __restrict__ hold, float* __restrict__ hnew) {
  int tid = blockIdx.x * blockDim.x + threadIdx.x;
  if (tid >= N_NODES * HID) return;
  int n = tid >> 7, c = tid & (HID - 1);
  const float* a = agg + (size_t)n * HEADS * HID + c;
  float s = (a[0] + a[HID] + a[2 * HID] + a[3 * HID]) * 0.25f + bconv[c];
  s = (s > 0.0f) ? s : 0.0f;
  hnew[tid] = s + hold[tid];
}

// ---------------- launch ----------------
extern "C" void kernel_launch(void* const* d_in, const int* in_sizes, int n_in,
                              void* d_out, int out_size, void* d_ws, size_t ws_size,
                              hipStream_t stream) {
  const float* x      = (const float*)d_in[0];
  const int*   ei     = (const int*)  d_in[1];
  const float* Win    = (const float*)d_in[2];
  const float* bin    = (const float*)d_in[3];
  const float* Wconv  = (const float*)d_in[4];   // [DEPTH][128][512]
  const float* attS   = (const float*)d_in[5];   // [DEPTH][4][128]
  const float* attD   = (const float*)d_in[6];
  const float* bconv  = (const float*)d_in[7];   // [DEPTH][128]
  const float* lng    = (const float*)d_in[8];
  const float* lnb    = (const float*)d_in[9];
  const float* Wout   = (const float*)d_in[10];  // [128][64]
  const float* bout   = (const float*)d_in[11];
  float* out = (float*)d_out;

  // workspace carve-up (floats)
  float* p   = (float*)d_ws;
  float* hA  = p;  p += (size_t)N_NODES * HID;            // current h
  float* hT  = p;  p += (size_t)N_NODES * HID;            // ln output / next h
  float* xh  = p;  p += (size_t)N_NODES * HEADS * HID;    // 102.4 MB (L2-resident)
  float* as_ = p;  p += (size_t)N_NODES * HEADS;
  float* ad_ = p;  p += (size_t)N_NODES * HEADS;
  float* m_  = p;  p += (size_t)N_NODES * HEADS;
  float* z_  = p;  p += (size_t)N_NODES * HEADS;
  float* ew  = p;  p += (size_t)E2 * HEADS;               // e then p, in place
  float* agg = p;  p += (size_t)N_NODES * HEADS * HID;    // 102.4 MB
  (void)ws_size; (void)n_in; (void)in_sizes; (void)out_size;

  const int TPB = 256;
  const int MT  = N_NODES / 16;                           // 3125, exact
  auto cdiv = [](long a, long b) { return (int)((a + b - 1) / b); };

  k_in_proj<<<cdiv((long)N_NODES * HID, TPB), TPB, 0, stream>>>(x, Win, bin, hA);

  for (int i = 0; i < DEPTH; ++i) {
    const float* Wc = Wconv + (size_t)i * HID * HEADS * HID;
    const float* as = attS  + (size_t)i * HEADS * HID;
    const float* ad = attD  + (size_t)i * HEADS * HID;
    const float* bc = bconv + (size_t)i * HID;
    const float* g  = lng   + (size_t)i * HID;
    const float* b  = lnb   + (size_t)i * HID;

    k_layernorm<<<cdiv(N_NODES, 8), TPB, 0, stream>>>(hA, hT, g, b);

    dim3 gg(cdiv(MT, 8), (HEADS * HID) / 64);             // 391 x 8
    k_gemm_wmma<<<gg, TPB, 0, stream>>>(hT, Wc, xh, nullptr, MT, HEADS * HID);

    k_alpha<<<cdiv((long)N_NODES * HEADS, 8), TPB, 0, stream>>>(xh, as, ad, as_, ad_);

    k_init<<<cdiv((long)N_NODES * HEADS * HID, TPB), TPB, 0, stream>>>(agg, m_, z_);

    k_edge_pass1<<<cdiv((long)E2 * HEADS, TPB), TPB, 0, stream>>>(ei, as_, ad_, ew, m_);
    k_edge_pass2<<<cdiv((long)E2 * HEADS, TPB), TPB, 0, stream>>>(ei, ew, m_, z_);
    k_edge_agg <<<cdiv((long)E2 * HEADS, 8),   TPB, 0, stream>>>(ei, ew, z_, xh, agg);

    k_finalize<<<cdiv((long)N_NODES * HID, TPB), TPB, 0, stream>>>(agg, bc, hA, hT);

    float* t = hA; hA = hT; hT = t;                       // swap
  }

  dim3 gf(cdiv(MT, 8), 1);
  k_gemm_wmma<<<gf, TPB, 0, stream>>>(hA, Wout, out, bout, MT, OUT_C);
}